// NeighborhoodSelfAttentionBlock_25503515803916
// MI455X (gfx1250) — compile-verified
//
#include <hip/hip_runtime.h>
#include <hip/hip_bf16.h>
#include <math.h>

// ---- problem constants (from reference) ----
#define D_MODEL 512
#define D_HEAD  64
#define NHEADS  8
#define CONDD   768
#define HH      32
#define WWID    32
#define NBATCH  2
#define TOKENS  (NBATCH*HH*WWID)   // 2048
#define QKVCOLS (3*D_MODEL)        // 1536
#define KWIN    7
#define NB      (KWIN*KWIN)        // 49
#define GK      512                // K dim of both big GEMMs
#define LDS_ROW 516                // 512 + 4 pad dwords (2 per 256) -> bank stride 4

typedef float        v2f  __attribute__((ext_vector_type(2)));
typedef float        v8f  __attribute__((ext_vector_type(8)));
typedef unsigned int u32x4 __attribute__((ext_vector_type(4)));
typedef int          i32x8 __attribute__((ext_vector_type(8)));
typedef int          i32x4 __attribute__((ext_vector_type(4)));

#if __has_builtin(__builtin_amdgcn_tensor_load_to_lds) && __has_builtin(__builtin_amdgcn_s_wait_tensorcnt)
#define HAS_TDM 1
#else
#define HAS_TDM 0
#endif

// LDS offset within a padded row: TDM pads 2 dwords after every 256 data dwords
__device__ __forceinline__ int lds_koff(int k) { return k + ((k >> 8) << 1); }

// ---------------------------------------------------------------------------
// Kernel 1: norm_scale[n][c] = dot(cond[n], w_cond[c]) + 1
// ---------------------------------------------------------------------------
__global__ void cond_gemm_kernel(const float* __restrict__ cond,
                                 const float* __restrict__ w_cond,
                                 float* __restrict__ ns)
{
    int idx = blockIdx.x * blockDim.x + threadIdx.x;
    if (idx >= NBATCH * D_MODEL) return;
    int n = idx / D_MODEL;
    int c = idx % D_MODEL;
    const float* cr = cond + (size_t)n * CONDD;
    const float* wr = w_cond + (size_t)c * CONDD;
    float s = 0.f;
    for (int k = 0; k < CONDD; ++k) s = fmaf(cr[k], wr[k], s);
    ns[idx] = s + 1.0f;
}

// ---------------------------------------------------------------------------
// Kernel 2: RMS-norm, one wave per token
// ---------------------------------------------------------------------------
__global__ __launch_bounds__(256)
void rmsnorm_kernel(const float* __restrict__ x,
                    const float* __restrict__ ns,
                    float* __restrict__ xn)
{
    int lane = threadIdx.x & 31;
    int t = blockIdx.x * (blockDim.x >> 5) + (threadIdx.x >> 5);
    if (t >= TOKENS) return;
    int n = t / (HH * WWID);
    const float* xr = x + (size_t)t * D_MODEL;
    const float* sc = ns + (size_t)n * D_MODEL;
    float v[16];
    float ss = 0.f;
#pragma unroll
    for (int i = 0; i < 16; ++i) {
        v[i] = xr[lane + i * 32];
        ss = fmaf(v[i], v[i], ss);
    }
#pragma unroll
    for (int m = 16; m >= 1; m >>= 1) ss += __shfl_xor(ss, m, 32);
    float inv = rsqrtf(ss * (1.0f / (float)D_MODEL) + 1e-6f);
    float* o = xn + (size_t)t * D_MODEL;
#pragma unroll
    for (int i = 0; i < 16; ++i)
        o[lane + i * 32] = v[i] * sc[lane + i * 32] * inv;
}

// ---------------------------------------------------------------------------
// Kernel 3/6: WMMA f32 GEMM  Y[M,N] = X[M,K=512] * Wt[N,K]^T (+ SKIP)
// Block (8 waves) owns one 16-row M tile staged in LDS via the Tensor Data
// Mover; each wave computes 16 x 64 of the output (NT=4 WMMA n-tiles).
// A fragments come from LDS (ds_load_b64), B from global, both register
// double-buffered across k-steps so WMMAs overlap loads.
// ---------------------------------------------------------------------------
template<int NT, bool HAS_SKIP>
__global__ __launch_bounds__(256)
void wmma_gemm_kernel(const float* __restrict__ X,
                      const float* __restrict__ Wt,
                      const float* __restrict__ SKIP,
                      float* __restrict__ Y,
                      int M, int N)
{
    __shared__ float sA[16 * LDS_ROW];        // 16 rows x 512 K (padded) = ~33KB

    const int nblocks = N >> 9;               // 512 output cols per block
    const int mt = blockIdx.x / nblocks;
    const int nbk = blockIdx.x - mt * nblocks;
    const int m0 = mt * 16;
    const int lane = threadIdx.x & 31;
    const int w = threadIdx.x >> 5;

    // ---- stage X[m0..m0+15][0..511] into LDS ----
#if HAS_TDM
    if (w == 0) {
        unsigned long long ga = (unsigned long long)(uintptr_t)(X + (size_t)m0 * GK);
        unsigned int ldsb = (unsigned int)(uintptr_t)(&sA[0]);   // low 32 bits = LDS offset
        // D# group0: count=1, lds_addr, 57b global_addr, type=2 ("image")
        u32x4 g0;
        g0[0] = 1u;
        g0[1] = ldsb;
        g0[2] = (unsigned int)ga;
        g0[3] = (unsigned int)(ga >> 32) | (2u << 30);
        // D# group1: data_size=4B(code2), pad_enable, pad_interval=256dw(code7),
        // pad_amount=2dw(code1); tensor 512 x M, tile 512 x 16, dim0 stride 512
        i32x8 g1 = {};
        g1[0] = (2 << 16) | (1 << 20) | (7 << 22) | (1 << 25);
        g1[1] = (GK & 0xffff) << 16;                     // tensor_dim0[15:0]
        g1[2] = ((GK >> 16) & 0xffff) | ((M & 0xffff) << 16); // dim0 hi | dim1 lo
        g1[3] = ((M >> 16) & 0xffff) | (GK << 16);       // dim1 hi | tile_dim0=512
        g1[4] = 16;                                      // tile_dim1=16, tile_dim2=0
        g1[5] = GK;                                      // tensor_dim0_stride lo32
        g1[6] = 0;
        g1[7] = 0;
        i32x4 gz4 = {};                                  // 2D tile: groups 2/3 zero
        i32x8 gz8 = {};
        __builtin_amdgcn_tensor_load_to_lds(g0, g1, gz4, gz4, gz8, 0);
        __builtin_amdgcn_s_wait_tensorcnt(0);
    }
#else
    for (int idx = threadIdx.x; idx < 16 * GK; idx += 256) {
        int r = idx >> 9, k = idx & (GK - 1);
        sA[r * LDS_ROW + lds_koff(k)] = X[(size_t)(m0 + r) * GK + k];
    }
#endif
    __syncthreads();

    const int n0 = (nbk << 9) + w * 64;       // this wave's first output column
    const int half = lane >> 4;               // 0: lanes 0-15, 1: lanes 16-31
    const int l16  = lane & 15;

    v8f acc[NT] = {};

    const float* aRow = sA + l16 * LDS_ROW + half * 2;   // A: (M=l16, K=k0+half*2..+1)
    const float* brow[NT];
#pragma unroll
    for (int t = 0; t < NT; ++t) {
        brow[t] = Wt + (size_t)(n0 + t * 16 + l16) * GK + half * 2;
        __builtin_prefetch(brow[t], 0, 3);    // global_prefetch_b8
    }

    // register double-buffer: fetch step k+1 while WMMAs consume step k
    v2f a_cur = *(const v2f*)(aRow + lds_koff(0));
    v2f b_cur[NT];
#pragma unroll
    for (int t = 0; t < NT; ++t) b_cur[t] = *(const v2f*)(brow[t]);

    for (int k0 = 0; k0 < GK; k0 += 4) {
        const int k1 = (k0 + 4) & (GK - 1);   // last iter wraps to 0 (dummy prefetch)
        v2f a_nxt = *(const v2f*)(aRow + lds_koff(k1));
        v2f b_nxt[NT];
#pragma unroll
        for (int t = 0; t < NT; ++t) b_nxt[t] = *(const v2f*)(brow[t] + k1);
#pragma unroll
        for (int t = 0; t < NT; ++t)
            acc[t] = __builtin_amdgcn_wmma_f32_16x16x4_f32(
                false, a_cur, false, b_cur[t], (short)0, acc[t], false, false);
        a_cur = a_nxt;
#pragma unroll
        for (int t = 0; t < NT; ++t) b_cur[t] = b_nxt[t];
    }

    // C/D layout: VGPR r -> (M = m0 + half*8 + r, N = n0 + t*16 + l16)
#pragma unroll
    for (int t = 0; t < NT; ++t) {
        int col = n0 + t * 16 + l16;
        const size_t base = (size_t)(m0 + half * 8) * N + col;
        if (HAS_SKIP) {
            float sk[8];
#pragma unroll
            for (int r = 0; r < 8; ++r) sk[r] = SKIP[base + (size_t)r * N];
#pragma unroll
            for (int r = 0; r < 8; ++r) acc[t][r] += sk[r];
        }
#pragma unroll
        for (int r = 0; r < 8; ++r)
            Y[base + (size_t)r * N] = acc[t][r];
    }
}

// ---------------------------------------------------------------------------
// Kernel 4: Q/K L2-norm (sqrt(scale)*rsqrt(sum q^2 + 1e-6)) + RoPE
// One wave per (token, head); lane owns dims {lane, lane+32}.
// ---------------------------------------------------------------------------
__global__ __launch_bounds__(256)
void qknorm_rope_kernel(float* __restrict__ qkv,
                        const float* __restrict__ pos,
                        const float* __restrict__ scale)
{
    int lane = threadIdx.x & 31;
    int wid = blockIdx.x * (blockDim.x >> 5) + (threadIdx.x >> 5);
    if (wid >= TOKENS * NHEADS) return;
    int t = wid / NHEADS;
    int h = wid - t * NHEADS;
    float p  = pos[t];
    float sq = sqrtf(scale[h]);
    int j = lane & 15;
    // freqs[h][j] = exp(log(pi) + (j*8+h)*ln(10)/128)
    float fr = expf(1.14472988585f + (float)(j * 8 + h) * 0.01798894546f);
    float th = p * fr;
    float c = cosf(th), s = sinf(th);
#pragma unroll
    for (int which = 0; which < 2; ++which) {   // 0 = q, 1 = k
        float* base = qkv + (size_t)t * QKVCOLS + which * D_MODEL + h * D_HEAD;
        float a = base[lane];        // dims 0..31
        float b = base[lane + 32];   // dims 32..63 (not rotated)
        float ss = fmaf(a, a, b * b);
#pragma unroll
        for (int m = 16; m >= 1; m >>= 1) ss += __shfl_xor(ss, m, 32);
        float g = sq * rsqrtf(ss + 1e-6f);
        a *= g; b *= g;
        // rotary on dims 0..31: x1 = dims 0..15 (lanes 0-15), x2 = dims 16..31
        float other = __shfl_xor(a, 16, 32);
        float na = (lane < 16) ? (a * c - other * s)   // x1' = x1*cos - x2*sin
                               : (a * c + other * s);  // x2' = x2*cos + x1*sin
        base[lane]      = na;
        base[lane + 32] = b;
    }
}

// ---------------------------------------------------------------------------
// Kernel 5: 7x7 neighborhood attention. One wave per (n, h, i, j).
// ---------------------------------------------------------------------------
__global__ __launch_bounds__(256)
void nb_attn_kernel(const float* __restrict__ qkv,
                    float* __restrict__ out)
{
    __shared__ float sc[8][NB + 3];
    int lane  = threadIdx.x & 31;
    int wslot = threadIdx.x >> 5;
    int wid = blockIdx.x * (blockDim.x >> 5) + wslot;
    if (wid >= NBATCH * NHEADS * HH * WWID) return;
    int j = wid % WWID;  int tmp = wid / WWID;
    int i = tmp % HH;    tmp /= HH;
    int h = tmp % NHEADS;
    int n = tmp / NHEADS;
    int t = (n * HH + i) * WWID + j;

    const float* qb = qkv + (size_t)t * QKVCOLS + h * D_HEAD;
    float q0 = qb[lane], q1 = qb[lane + 32];

    int i0 = min(max(i - KWIN / 2, 0), HH - KWIN);
    int j0 = min(max(j - KWIN / 2, 0), WWID - KWIN);

    float mx = -1e30f;
    for (int kk = 0; kk < NB; ++kk) {
        int ki = i0 + kk / KWIN, kj = j0 + kk % KWIN;
        int kt = (n * HH + ki) * WWID + kj;
        const float* kb = qkv + (size_t)kt * QKVCOLS + D_MODEL + h * D_HEAD;
        float part = fmaf(q0, kb[lane], q1 * kb[lane + 32]);
#pragma unroll
        for (int m = 16; m >= 1; m >>= 1) part += __shfl_xor(part, m, 32);
        if (lane == 0) sc[wslot][kk] = part;
        mx = fmaxf(mx, part);
    }
    // softmax over 49 scores (per-wave LDS ordering handled by DScnt waits)
    float esum = 0.f;
    for (int kk = lane; kk < NB; kk += 32) {
        float e = expf(sc[wslot][kk] - mx);
        sc[wslot][kk] = e;
        esum += e;
    }
#pragma unroll
    for (int m = 16; m >= 1; m >>= 1) esum += __shfl_xor(esum, m, 32);
    float invs = 1.0f / esum;

    float o0 = 0.f, o1 = 0.f;
    for (int kk = 0; kk < NB; ++kk) {
        int ki = i0 + kk / KWIN, kj = j0 + kk % KWIN;
        int kt = (n * HH + ki) * WWID + kj;
        const float* vb = qkv + (size_t)kt * QKVCOLS + 2 * D_MODEL + h * D_HEAD;
        float a = sc[wslot][kk] * invs;
        o0 = fmaf(a, vb[lane], o0);
        o1 = fmaf(a, vb[lane + 32], o1);
    }
    float* ob = out + (size_t)t * D_MODEL + h * D_HEAD;
    ob[lane]      = o0;
    ob[lane + 32] = o1;
}

// ---------------------------------------------------------------------------
extern "C" void kernel_launch(void* const* d_in, const int* in_sizes, int n_in,
                              void* d_out, int out_size, void* d_ws, size_t ws_size,
                              hipStream_t stream)
{
    (void)in_sizes; (void)n_in; (void)out_size; (void)ws_size;
    const float* x      = (const float*)d_in[0];
    const float* pos    = (const float*)d_in[1];
    const float* cond   = (const float*)d_in[2];
    const float* w_cond = (const float*)d_in[3];
    const float* w_qkv  = (const float*)d_in[4];
    const float* scale  = (const float*)d_in[5];
    const float* w_out  = (const float*)d_in[6];
    float* out = (float*)d_out;

    float* ws  = (float*)d_ws;
    float* ns  = ws;                                  // 1024 floats
    float* xn  = ns + NBATCH * D_MODEL;               // TOKENS*D_MODEL (reused for attn out)
    float* qkv = xn + (size_t)TOKENS * D_MODEL;       // TOKENS*1536

    cond_gemm_kernel<<<4, 256, 0, stream>>>(cond, w_cond, ns);
    rmsnorm_kernel<<<TOKENS / 8, 256, 0, stream>>>(x, ns, xn);

    {   // QKV projection: 2048 x 512 x 1536 ; 3 blocks of 512 cols per M tile
        int blocks = (TOKENS / 16) * (QKVCOLS >> 9);  // 128 * 3 = 384
        wmma_gemm_kernel<4, false><<<blocks, 256, 0, stream>>>(
            xn, w_qkv, nullptr, qkv, TOKENS, QKVCOLS);
    }

    qknorm_rope_kernel<<<(TOKENS * NHEADS) / 8, 256, 0, stream>>>(qkv, pos, scale);
    nb_attn_kernel<<<(TOKENS * NHEADS) / 8, 256, 0, stream>>>(qkv, xn);

    {   // Output projection + residual: 2048 x 512 x 512 ; 1 block per M tile
        int blocks = (TOKENS / 16) * (D_MODEL >> 9);  // 128
        wmma_gemm_kernel<4, true><<<blocks, 256, 0, stream>>>(
            xn, w_out, x, out, TOKENS, D_MODEL);
    }
}